// LambdaConv_15547781612043
// MI455X (gfx1250) — compile-verified
//
#include <hip/hip_runtime.h>

typedef __attribute__((ext_vector_type(2))) float v2f;
typedef __attribute__((ext_vector_type(8))) float v8f;

#define DIMC   256
#define C_OUT  144
#define NPIX   65536      // b*h*w = 16*64*64
#define MPB    4096       // h*w per batch
#define DIMK   16
#define DIMV   64
#define SCOPE  23
#define TAPS   (SCOPE*SCOPE)   // 529
#define PADR   11
#define EPSV   1e-5f

#define PATCH_H 23
#define PATCH_W 38
#define PATCH_P 40        // padded row stride (floats)

// workspace layout (floats)
#define QKV_OFF   0
#define QKV_SZ    ((long)C_OUT * NPIX)
#define STAT_OFF  (QKV_OFF + QKV_SZ)
#define STAT_SZ   (2 * C_OUT)
#define LAM_OFF   (STAT_OFF + STAT_SZ)

// LDS byte offset (wave-relative) from a generic pointer to __shared__ data:
// generic shared addresses are {SHARED_BASE, offset32}; low 32 bits are the
// LDS address that DS / async-to-LDS instructions expect in their vdst VGPR operand.
__device__ __forceinline__ unsigned lds_off(const void* p) {
  return (unsigned)(unsigned long long)(uintptr_t)p;
}
__device__ __forceinline__ void async_load_b32(unsigned lds, const void* g) {
  asm volatile("global_load_async_to_lds_b32 %0, %1, off"
               :: "v"(lds), "v"((unsigned long long)(uintptr_t)g) : "memory");
}
__device__ __forceinline__ void async_load_b128(unsigned lds, const void* g) {
  asm volatile("global_load_async_to_lds_b128 %0, %1, off"
               :: "v"(lds), "v"((unsigned long long)(uintptr_t)g) : "memory");
}
__device__ __forceinline__ void wait_async0() {
  asm volatile("s_wait_asynccnt 0x0" ::: "memory");
}

// ---------------------------------------------------------------------------
// K1: qkv = w_qkv(144x256) @ x  over 65536 pixel columns.  fp32 WMMA 16x16x4.
// Memory-bound (67MB of x); one wave per 16x16 output tile.
// ---------------------------------------------------------------------------
__global__ __launch_bounds__(256) void qkv_gemm(const float* __restrict__ x,
                                                const float* __restrict__ w,
                                                float* __restrict__ qkv) {
  const int lane = threadIdx.x & 31;
  const int wave = threadIdx.x >> 5;
  const int tile = blockIdx.x * 8 + wave;          // 36864 tiles
  const int mt = tile % 9;                         // 9 * 16 = 144 rows
  const int nt = tile / 9;                         // 4096 * 16 = 65536 cols
  const int o0 = mt * 16;
  const int p0 = nt * 16;
  const int l  = lane & 15;
  const bool hi = lane >= 16;

  const int p = p0 + l;                            // column (pixel) index
  const long colBase = (long)(p >> 12) * DIMC * MPB + (p & (MPB - 1));
  const int orow = o0 + l;

  v8f acc = {};
  #pragma unroll 4
  for (int k0 = 0; k0 < DIMC; k0 += 4) {
    if (k0 + 8 < DIMC)
      __builtin_prefetch(&x[colBase + (long)(k0 + 8) * MPB], 0, 1);
    v2f a  = *(const v2f*)(w + orow * DIMC + k0 + (hi ? 2 : 0));  // b64 load
    v2f bf;
    bf.x = x[colBase + (long)(k0 + (hi ? 2 : 0)) * MPB];
    bf.y = x[colBase + (long)(k0 + (hi ? 3 : 1)) * MPB];
    acc = __builtin_amdgcn_wmma_f32_16x16x4_f32(false, a, false, bf,
                                                (short)0, acc, false, false);
  }
  #pragma unroll
  for (int r = 0; r < 8; ++r) {
    const int row = o0 + r + (hi ? 8 : 0);
    qkv[(long)row * NPIX + p] = acc[r];
  }
}

// ---------------------------------------------------------------------------
// K2: BatchNorm statistics for q channels (0..63) and v channels (80..143).
// ---------------------------------------------------------------------------
__global__ __launch_bounds__(256) void bn_stats(const float* __restrict__ qkv,
                                                float* __restrict__ stats) {
  const int ch = (blockIdx.x < 64) ? blockIdx.x : blockIdx.x + 16;
  __shared__ float sS[256];
  __shared__ float sQ2[256];
  float s = 0.f, sq = 0.f;
  const float* row = qkv + (long)ch * NPIX;
  for (int i = threadIdx.x; i < NPIX; i += 256) {
    float v = row[i];
    s += v; sq += v * v;
  }
  sS[threadIdx.x] = s; sQ2[threadIdx.x] = sq;
  __syncthreads();
  for (int off = 128; off > 0; off >>= 1) {
    if (threadIdx.x < off) {
      sS[threadIdx.x]  += sS[threadIdx.x + off];
      sQ2[threadIdx.x] += sQ2[threadIdx.x + off];
    }
    __syncthreads();
  }
  if (threadIdx.x == 0) {
    float mean = sS[0] * (1.f / NPIX);
    float var  = sQ2[0] * (1.f / NPIX) - mean * mean;
    stats[ch * 2]     = mean;
    stats[ch * 2 + 1] = rsqrtf(var + EPSV);
  }
}

// ---------------------------------------------------------------------------
// K3: apply BN (in-place on qkv workspace) to q and v channels.
// ---------------------------------------------------------------------------
__global__ __launch_bounds__(256) void bn_apply(float* __restrict__ qkv,
                                                const float* __restrict__ stats,
                                                const float* __restrict__ gq,
                                                const float* __restrict__ bq,
                                                const float* __restrict__ gv,
                                                const float* __restrict__ bv) {
  long idx = (long)blockIdx.x * 256 + threadIdx.x;   // over 128 * NPIX
  int  ci  = (int)(idx >> 16);                        // /NPIX
  long p   = idx & (NPIX - 1);
  int  ch  = (ci < 64) ? ci : ci + 16;
  float g, bb;
  if (ci < 64) { g = gq[ci];      bb = bq[ci]; }
  else         { g = gv[ci - 64]; bb = bv[ci - 64]; }
  float mean = stats[ch * 2], inv = stats[ch * 2 + 1];
  long a = (long)ch * NPIX + p;
  qkv[a] = (qkv[a] - mean) * inv * g + bb;
}

// ---------------------------------------------------------------------------
// K4: softmax over m=4096 for each of the 16 k-channels per batch (in-place).
// ---------------------------------------------------------------------------
__global__ __launch_bounds__(256) void k_softmax(float* __restrict__ qkv) {
  const int b  = blockIdx.x >> 4;
  const int kc = blockIdx.x & 15;
  float* row = qkv + (long)(64 + kc) * NPIX + (long)b * MPB;
  __shared__ float red[256];
  float m = -3.4e38f;
  for (int i = threadIdx.x; i < MPB; i += 256) m = fmaxf(m, row[i]);
  red[threadIdx.x] = m; __syncthreads();
  for (int off = 128; off > 0; off >>= 1) {
    if (threadIdx.x < off)
      red[threadIdx.x] = fmaxf(red[threadIdx.x], red[threadIdx.x + off]);
    __syncthreads();
  }
  m = red[0];
  __syncthreads();
  float s = 0.f;
  for (int i = threadIdx.x; i < MPB; i += 256) s += __expf(row[i] - m);
  red[threadIdx.x] = s; __syncthreads();
  for (int off = 128; off > 0; off >>= 1) {
    if (threadIdx.x < off) red[threadIdx.x] += red[threadIdx.x + off];
    __syncthreads();
  }
  float inv = 1.f / red[0];
  for (int i = threadIdx.x; i < MPB; i += 256) row[i] = __expf(row[i] - m) * inv;
}

// ---------------------------------------------------------------------------
// K5: content_lambda[b,k,v] = sum_m softmax_k[b,k,m] * v_bn[b,v,m]
// One block per batch; each wave = one 16-wide v tile; K loop over m=4096.
// ---------------------------------------------------------------------------
__global__ __launch_bounds__(128) void content_lambda(const float* __restrict__ qkv,
                                                      float* __restrict__ lam) {
  const int b    = blockIdx.x;
  const int wave = threadIdx.x >> 5;
  const int lane = threadIdx.x & 31;
  const int l    = lane & 15;
  const bool hi  = lane >= 16;
  const int v0   = wave * 16;
  const float* K = qkv + (long)64 * NPIX + (long)b * MPB;
  const float* V = qkv + (long)80 * NPIX + (long)b * MPB;
  v8f acc = {};
  for (int m0 = 0; m0 < MPB; m0 += 4) {
    v2f a  = *(const v2f*)(K + (long)l * NPIX + m0 + (hi ? 2 : 0));        // b64
    v2f bf = *(const v2f*)(V + (long)(v0 + l) * NPIX + m0 + (hi ? 2 : 0)); // b64
    acc = __builtin_amdgcn_wmma_f32_16x16x4_f32(false, a, false, bf,
                                                (short)0, acc, false, false);
  }
  #pragma unroll
  for (int r = 0; r < 8; ++r) {
    int kk = r + (hi ? 8 : 0);
    lam[(long)b * (DIMK * DIMV) + kk * DIMV + (v0 + l)] = acc[r];
  }
}

// ---------------------------------------------------------------------------
// K6: fused position path + output.
// Per block: one batch b, one row-tile of 16 consecutive pixels.
//   * conv weights (16x529, 33KB) staged once per block via
//     global_load_async_to_lds_b128 (ASYNCcnt path)
//   * per (wave, v): 23x38 v-patch staged via global_load_async_to_lds_b32,
//     boundary zeros via disjoint ds_store (no overlap -> no ordering hazard)
//   * PL tile (16k x 16px) accumulated with fp32 WMMA over 529 taps,
//     accumulator pre-seeded with content lambda[k,v] + b_lconv[k]
//   * PL spilled to LDS and contracted with the q tile -> out[n,v,m]
// The 268MB position_lambda tensor is never materialized.
// ---------------------------------------------------------------------------
__global__ __launch_bounds__(128) void lambda_out(const float* __restrict__ qkv,
                                                  const float* __restrict__ lam,
                                                  const float* __restrict__ wl,
                                                  const float* __restrict__ bl,
                                                  float* __restrict__ out) {
  const int t     = blockIdx.x;       // 4096 tiles
  const int b     = t >> 8;
  const int mbase = (t & 255) * 16;   // 16 pixels, same image row
  const int lane  = threadIdx.x & 31;
  const int wave  = threadIdx.x >> 5;
  const int l     = lane & 15;
  const bool hi   = lane >= 16;
  const int h0    = mbase >> 6;
  const int w0    = mbase & 63;

  __shared__ float sQ[64 * 16];                    // q channels x pixels
  __shared__ float sPL[4][16 * 16];                // per-wave PL tile (k x px)
  __shared__ float sWl[DIMK * TAPS];               // conv weights, 16x529
  __shared__ float sPatch[4][PATCH_H * PATCH_P];   // per-wave v patch

  // ---- stage conv weights (2116 x b128) + q tile (256 x b128), async ----
  for (int i = threadIdx.x; i < (DIMK * TAPS) / 4; i += 128)
    async_load_b128(lds_off(&sWl[i * 4]), wl + i * 4);
  for (int i = threadIdx.x; i < 256; i += 128) {
    const float* gp = qkv + (long)(i >> 2) * NPIX + (long)b * MPB + mbase + (i & 3) * 4;
    async_load_b128(lds_off(&sQ[i * 4]), gp);
  }
  wait_async0();
  __syncthreads();

  const float* Vb = qkv + (long)80 * NPIX + (long)b * MPB;
  const int y0 = h0 - PADR;
  const int x0 = w0 - PADR;

  for (int vbase = 0; vbase < DIMV; vbase += 4) {
    const int v = vbase + wave;
    float* pw = &sPatch[wave][0];
    const float* Vrow = Vb + (long)v * NPIX;

    // ---- stage 23x38 patch for this v (async; zeros for out-of-image) ----
    for (int e = lane; e < PATCH_H * PATCH_W; e += 32) {
      int py = e / PATCH_W, px = e % PATCH_W;
      int y = y0 + py, xg = x0 + px;
      unsigned dst = lds_off(&pw[py * PATCH_P + px]);
      if ((unsigned)y < 64u && (unsigned)xg < 64u)
        async_load_b32(dst, Vrow + y * 64 + xg);
      else
        pw[py * PATCH_P + px] = 0.f;
    }
    wait_async0();

    // ---- accumulator: lambda[k,v] + bias[k] ----
    v8f acc;
    #pragma unroll
    for (int r = 0; r < 8; ++r) {
      int kk = r + (hi ? 8 : 0);
      acc[r] = lam[(long)b * (DIMK * DIMV) + kk * DIMV + v] + bl[kk];
    }

    // ---- conv via WMMA over 529 taps (A, B from LDS) ----
    for (int s = 0; s < TAPS; s += 4) {           // 133 WMMA per (v, tile)
      int s0 = s + (hi ? 2 : 0);
      int s1 = s0 + 1;
      v2f a, bf;
      a.x  = (s0 < TAPS) ? sWl[l * TAPS + s0] : 0.f;
      a.y  = (s1 < TAPS) ? sWl[l * TAPS + s1] : 0.f;
      bf.x = (s0 < TAPS) ? pw[(s0 / SCOPE) * PATCH_P + (s0 % SCOPE) + l] : 0.f;
      bf.y = (s1 < TAPS) ? pw[(s1 / SCOPE) * PATCH_P + (s1 % SCOPE) + l] : 0.f;
      acc = __builtin_amdgcn_wmma_f32_16x16x4_f32(false, a, false, bf,
                                                  (short)0, acc, false, false);
    }

    // ---- spill PL tile, contract with q ----
    #pragma unroll
    for (int r = 0; r < 8; ++r) {
      int kk = r + (hi ? 8 : 0);
      sPL[wave][kk * 16 + l] = acc[r];
    }
    __syncthreads();
    #pragma unroll
    for (int e = 0; e < 2; ++e) {
      int idx = lane * 2 + e;       // 0..63
      int n   = idx & 3;
      int px  = idx >> 2;
      float o = 0.f;
      #pragma unroll
      for (int kk = 0; kk < DIMK; ++kk)
        o += sQ[(kk * 4 + n) * 16 + px] * sPL[wave][kk * 16 + px];
      int oc = n * DIMV + v;
      out[((long)b * 256 + oc) * MPB + mbase + px] = o;
    }
    __syncthreads();
  }
}

// ---------------------------------------------------------------------------
extern "C" void kernel_launch(void* const* d_in, const int* in_sizes, int n_in,
                              void* d_out, int out_size, void* d_ws, size_t ws_size,
                              hipStream_t stream) {
  (void)in_sizes; (void)n_in; (void)out_size; (void)ws_size;
  const float* x     = (const float*)d_in[0];
  const float* w_qkv = (const float*)d_in[1];
  const float* gq    = (const float*)d_in[2];
  const float* bq    = (const float*)d_in[3];
  const float* gv    = (const float*)d_in[4];
  const float* bv    = (const float*)d_in[5];
  const float* wl    = (const float*)d_in[6];
  const float* bl    = (const float*)d_in[7];
  float* out = (float*)d_out;
  float* ws  = (float*)d_ws;

  float* qkv   = ws + QKV_OFF;
  float* stats = ws + STAT_OFF;
  float* lamb  = ws + LAM_OFF;

  qkv_gemm<<<4608, 256, 0, stream>>>(x, w_qkv, qkv);
  bn_stats<<<128, 256, 0, stream>>>(qkv, stats);
  bn_apply<<<32768, 256, 0, stream>>>(qkv, stats, gq, bq, gv, bv);
  k_softmax<<<256, 256, 0, stream>>>(qkv);
  content_lambda<<<16, 128, 0, stream>>>(qkv, lamb);
  lambda_out<<<4096, 128, 0, stream>>>(qkv, lamb, wl, bl, out);
}